// SemanticDecoder_11811160064302
// MI455X (gfx1250) — compile-verified
//
#include <hip/hip_runtime.h>

typedef __attribute__((ext_vector_type(2))) float v2f;
typedef __attribute__((ext_vector_type(8))) float v8f;

#define B_ROWS 65536
#define HASHLEN 64
#define D_COLS 300

// One wave computes one 16x16 tile of BOTH GEMMs (mean and sigma), fused with
// the elementwise epilogue. f32 WMMA (16x16x4) keeps exact f32 accumulation.
//
// Bandwidth-bound problem (~331 MB mandatory HBM traffic, ~14 us at 23.3 TB/s):
//  - activation b (16.8 MB) is L2-resident across the 19 column-tile re-reads
//  - eps/outputs are streamed with non-temporal hints
// The k-loop is branch-free: out-of-range weight columns are handled by
// clamping the address (garbage only reaches output columns that are never
// stored), so EXEC stays all-1s and all loads use immediate offsets.
__global__ __launch_bounds__(256) void semantic_decoder_wmma(
    const float* __restrict__ b,     // [65536, 64]
    const float* __restrict__ Wmu,   // [64, 300]
    const float* __restrict__ bmu,   // [300]
    const float* __restrict__ Wsig,  // [64, 300]
    const float* __restrict__ bsig,  // [300]
    const float* __restrict__ eps,   // [65536, 300]
    float* __restrict__ out)         // [3, 65536, 300] = sample, mean, var
{
    const int lane = threadIdx.x & 31;
    const int wave = threadIdx.x >> 5;

    const int mt = blockIdx.y * 8 + wave;   // row tile: 0..4095
    const int nt = blockIdx.x;              // col tile: 0..18
    const int row0 = mt * 16;
    const int col0 = nt * 16;

    // WMMA f32 fragment geometry (wave32):
    //   A 16x4: lane L holds A[L&15][kh], A[L&15][kh+1], kh = 2*(L>=16)
    //   B 4x16: lane L holds B[kh][L&15], B[kh+1][L&15]
    const int r  = lane & 15;
    const int kh = (lane >> 4) << 1;

    const int col  = col0 + r;
    // Clamped column keeps every weight/bias/eps/output address in-bounds.
    // Lanes with col >= D_COLS compute garbage that is never stored.
    const int colC = (col < D_COLS) ? col : (D_COLS - 1);

    // Fixed base pointers -> all loop loads use immediate offsets.
    const float* __restrict__ arowk = b + (size_t)(row0 + r) * HASHLEN + kh;
    const float* __restrict__ wmu   = Wmu  + (size_t)kh * D_COLS + colC;
    const float* __restrict__ wsg   = Wsig + (size_t)kh * D_COLS + colC;

    v8f accMu = {};
    v8f accSg = {};

#pragma unroll
    for (int k0 = 0; k0 < HASHLEN; k0 += 4) {
        v2f a;
        a.x = arowk[k0];
        a.y = arowk[k0 + 1];

        v2f bm;
        bm.x = wmu[k0 * D_COLS];
        bm.y = wmu[k0 * D_COLS + D_COLS];

        v2f bs;
        bs.x = wsg[k0 * D_COLS];
        bs.y = wsg[k0 * D_COLS + D_COLS];

        accMu = __builtin_amdgcn_wmma_f32_16x16x4_f32(
            false, a, false, bm, (short)0, accMu, false, false);
        accSg = __builtin_amdgcn_wmma_f32_16x16x4_f32(
            false, a, false, bs, (short)0, accSg, false, false);
    }

    const float biasMu = bmu[colC];
    const float biasSg = bsig[colC];

    // C/D layout: VGPR v, lanes 0-15 -> M=v; lanes 16-31 -> M=v+8; N=lane&15
    const int rbase = row0 + ((lane >> 4) << 3);
    const size_t base = (size_t)rbase * D_COLS + colC;

    const float* __restrict__ epsp      = eps + base;
    float* __restrict__ outSample = out + base;
    float* __restrict__ outMean   = out + (size_t)B_ROWS * D_COLS + base;
    float* __restrict__ outVar    = out + 2 * (size_t)B_ROWS * D_COLS + base;

    if (col0 + 16 <= D_COLS) {
        // Full tile (18 of 19 column tiles): no lane predication at all.
#pragma unroll
        for (int v = 0; v < 8; ++v) {
            const size_t o = (size_t)v * D_COLS;
            const float m   = accMu[v] + biasMu;
            const float s   = accSg[v] + biasSg;
            const float var = s * s;
            const float e   = __builtin_nontemporal_load(epsp + o);
            const float smp = fmaf(fabsf(s), e, m);     // sqrt(s*s) == |s|
            __builtin_nontemporal_store(smp, outSample + o);
            __builtin_nontemporal_store(m,   outMean   + o);
            __builtin_nontemporal_store(var, outVar    + o);
        }
    } else if (col < D_COLS) {
        // Tail tile: predicate once for the whole store loop.
#pragma unroll
        for (int v = 0; v < 8; ++v) {
            const size_t o = (size_t)v * D_COLS;
            const float m   = accMu[v] + biasMu;
            const float s   = accSg[v] + biasSg;
            const float var = s * s;
            const float e   = __builtin_nontemporal_load(epsp + o);
            const float smp = fmaf(fabsf(s), e, m);
            __builtin_nontemporal_store(smp, outSample + o);
            __builtin_nontemporal_store(m,   outMean   + o);
            __builtin_nontemporal_store(var, outVar    + o);
        }
    }
}

extern "C" void kernel_launch(void* const* d_in, const int* in_sizes, int n_in,
                              void* d_out, int out_size, void* d_ws, size_t ws_size,
                              hipStream_t stream) {
    const float* b    = (const float*)d_in[0];
    // d_in[1] = labels (int64), unused by the reference computation
    const float* Wmu  = (const float*)d_in[2];
    const float* bmu  = (const float*)d_in[3];
    const float* Wsig = (const float*)d_in[4];
    const float* bsig = (const float*)d_in[5];
    const float* eps  = (const float*)d_in[6];
    float* out = (float*)d_out;

    // grid.x = 19 column tiles (ceil(300/16)); grid.y = 512 blocks of
    // 8 waves x 16 rows = 128 rows each -> 65536 rows.
    dim3 grid((D_COLS + 15) / 16, B_ROWS / 128);
    semantic_decoder_wmma<<<grid, 256, 0, stream>>>(b, Wmu, bmu, Wsig, bsig, eps, out);
}